// GIN_42949672960222
// MI455X (gfx1250) — compile-verified
//
#include <hip/hip_runtime.h>

// GIN forward for MI455X (gfx1250, wave32).
// d_in layout (setup_inputs dict order, flattened):
//  [0]=x (N*128 f32) [1]=src (E i32) [2]=dst (E i32)
//  per layer L (base=3+9L): W1,b1,g1,be1,W2,b2,g2,be2,eps
//  [30]=Wh (128*64) [31]=bh (64)
// d_out = logits (N*64) ++ h (N*128), f32.

#define NNODES 100000
#define FEAT   128
#define OUTF   64

typedef __attribute__((ext_vector_type(2))) float v2f;
typedef __attribute__((ext_vector_type(8))) float v8f;

// ---------------------------------------------------------------------------
// Edge scatter-add: agg[dst[e]][:] += h[src[e]][:]
// One wave32 per edge; lane owns features [4*lane, 4*lane+4) -> each wave
// touches one contiguous 512B row on both the gather and the atomic side.
// ---------------------------------------------------------------------------
__global__ __launch_bounds__(256)
void scatter_add_kernel(const float* __restrict__ h,
                        const int* __restrict__ src,
                        const int* __restrict__ dst,
                        float* __restrict__ agg, int E)
{
    int t = blockIdx.x * 256 + threadIdx.x;
    int e = t >> 5;
    if (e >= E) return;
    int lane = t & 31;
    int s = src[e];
    int d = dst[e];
    float4 v = *(const float4*)(h + (size_t)s * FEAT + lane * 4);
    float* p = agg + (size_t)d * FEAT + lane * 4;
    atomicAdd(p + 0, v.x);
    atomicAdd(p + 1, v.y);
    atomicAdd(p + 2, v.z);
    atomicAdd(p + 3, v.w);
}

// ---------------------------------------------------------------------------
// GEMM: Y[N x NOUT] = A[N x 128] * W[128 x NOUT] + bias, f32 WMMA 16x16x4.
// FUSE:  A = (1+eps)*h + agg  (GIN combine fused into the A-operand load)
// STATS: accumulate per-column sum / sumsq (for BatchNorm) via LDS->global.
// Block = 8 waves; wave handles a 16-row strip across all N-tiles.
// W staged in LDS (<=64KB of the 320KB WGP pool).
// ---------------------------------------------------------------------------
template<int NOUT, bool FUSE, bool STATS>
__global__ __launch_bounds__(256)
void gemm_wmma_kernel(const float* __restrict__ Ah,
                      const float* __restrict__ Aagg,
                      const float* __restrict__ epsPtr,
                      const float* __restrict__ W,
                      const float* __restrict__ bias,
                      float* __restrict__ Y,
                      float* __restrict__ stats,
                      int nRows)
{
    __shared__ float sW[FEAT * NOUT];
    __shared__ float sStats[2 * NOUT];

    const int tid = threadIdx.x;

    // Stage W into LDS (coalesced float4).
    for (int i = tid * 4; i < FEAT * NOUT; i += 256 * 4)
        *(float4*)(sW + i) = *(const float4*)(W + i);
    if (STATS)
        for (int t2 = tid; t2 < 2 * NOUT; t2 += 256) sStats[t2] = 0.0f;
    __syncthreads();

    const int wave = tid >> 5;
    const int lane = tid & 31;
    const int half = lane >> 4;   // 0: lanes 0-15, 1: lanes 16-31
    const int l16  = lane & 15;
    const int strip = blockIdx.x * 8 + wave;
    const int nStrips = nRows >> 4;

    if (strip < nStrips) {
        const int row0 = strip * 16;
        const int m = row0 + l16;                  // A row for this lane
        float scale = 1.0f;
        if (FUSE) scale = 1.0f + epsPtr[0];
        const float* rh = Ah + (size_t)m * FEAT;
        const float* ra = FUSE ? (Aagg + (size_t)m * FEAT) : nullptr;

        v8f acc[NOUT / 16];
        #pragma unroll
        for (int j = 0; j < NOUT / 16; ++j) {
            v8f z = {0.f, 0.f, 0.f, 0.f, 0.f, 0.f, 0.f, 0.f};
            acc[j] = z;
        }

        // K loop: 32 steps of K=4.
        #pragma unroll 4
        for (int k4 = 0; k4 < FEAT; k4 += 4) {
            const int kk = k4 + 2 * half;          // A: VGPR0->K=kk, VGPR1->K=kk+1
            float2 hv = *(const float2*)(rh + kk);
            v2f a;
            if (FUSE) {
                float2 av = *(const float2*)(ra + kk);
                a.x = scale * hv.x + av.x;
                a.y = scale * hv.y + av.y;
            } else {
                a.x = hv.x;
                a.y = hv.y;
            }
            #pragma unroll
            for (int j = 0; j < NOUT / 16; ++j) {
                v2f b;                              // B mirrored layout from LDS
                b.x = sW[kk * NOUT + j * 16 + l16];
                b.y = sW[(kk + 1) * NOUT + j * 16 + l16];
                acc[j] = __builtin_amdgcn_wmma_f32_16x16x4_f32(
                    false, a, false, b, (short)0, acc[j], false, false);
            }
        }

        // Epilogue: bias, store, BN-stat partials.
        #pragma unroll
        for (int j = 0; j < NOUT / 16; ++j) {
            const int n = j * 16 + l16;
            const float bn = bias[n];
            float psum = 0.0f, psq = 0.0f;
            #pragma unroll
            for (int i = 0; i < 8; ++i) {
                float v = acc[j][i] + bn;           // D: M = row0 + i + 8*half
                const int mm = row0 + i + 8 * half;
                Y[(size_t)mm * NOUT + n] = v;
                psum += v;
                psq  += v * v;
            }
            if (STATS) {
                atomicAdd(&sStats[n], psum);        // ds_add_f32
                atomicAdd(&sStats[NOUT + n], psq);
            }
        }
    }

    if (STATS) {
        __syncthreads();
        for (int t2 = tid; t2 < 2 * NOUT; t2 += 256)
            atomicAdd(&stats[t2], sStats[t2]);      // one global atomic/col/WG
    }
}

// ---------------------------------------------------------------------------
// BatchNorm (training mode, population var) + optional ReLU, in place.
// ---------------------------------------------------------------------------
template<bool RELU>
__global__ __launch_bounds__(256)
void bn_kernel(float* __restrict__ y,
               const float* __restrict__ stats,
               const float* __restrict__ g,
               const float* __restrict__ be,
               int total, float invN)
{
    int idx = blockIdx.x * 256 + threadIdx.x;
    int stride = gridDim.x * 256;
    for (; idx < total; idx += stride) {
        int n = idx & (FEAT - 1);
        float mean = stats[n] * invN;
        float var  = stats[FEAT + n] * invN - mean * mean;
        float v = y[idx];
        v = g[n] * (v - mean) * rsqrtf(var + 1e-5f) + be[n];
        if (RELU) v = fmaxf(v, 0.0f);
        y[idx] = v;
    }
}

// ---------------------------------------------------------------------------
extern "C" void kernel_launch(void* const* d_in, const int* in_sizes, int n_in,
                              void* d_out, int out_size, void* d_ws, size_t ws_size,
                              hipStream_t stream)
{
    (void)n_in; (void)out_size; (void)ws_size;
    const float* x   = (const float*)d_in[0];
    const int*   src = (const int*)d_in[1];
    const int*   dst = (const int*)d_in[2];
    const int    E   = in_sizes[1];

    // Workspace: 3 ping buffers of N*128 f32 + two 256-float stat blocks.
    float* bufH   = (float*)d_ws;
    float* bufAgg = bufH   + (size_t)NNODES * FEAT;
    float* bufY   = bufAgg + (size_t)NNODES * FEAT;
    float* stats1 = bufY   + (size_t)NNODES * FEAT;
    float* stats2 = stats1 + 2 * FEAT;

    const int scatterBlocks = (E * 32 + 255) / 256;
    const int gemmBlocks    = ((NNODES / 16) + 7) / 8;          // 782
    const int bnBlocks      = (NNODES * FEAT + 255) / 256;
    const float invN        = 1.0f / (float)NNODES;

    const float* hIn = x;
    for (int L = 0; L < 3; ++L) {
        const int base = 3 + L * 9;
        const float* W1  = (const float*)d_in[base + 0];
        const float* b1  = (const float*)d_in[base + 1];
        const float* g1  = (const float*)d_in[base + 2];
        const float* be1 = (const float*)d_in[base + 3];
        const float* W2  = (const float*)d_in[base + 4];
        const float* b2  = (const float*)d_in[base + 5];
        const float* g2  = (const float*)d_in[base + 6];
        const float* be2 = (const float*)d_in[base + 7];
        const float* eps = (const float*)d_in[base + 8];

        hipMemsetAsync(bufAgg, 0, (size_t)NNODES * FEAT * sizeof(float), stream);
        hipMemsetAsync(stats1, 0, 2 * FEAT * sizeof(float), stream);
        hipMemsetAsync(stats2, 0, 2 * FEAT * sizeof(float), stream);

        scatter_add_kernel<<<scatterBlocks, 256, 0, stream>>>(hIn, src, dst, bufAgg, E);

        // Y1 = ((1+eps)h + agg) @ W1 + b1, with BN stat accumulation.
        gemm_wmma_kernel<FEAT, true, true><<<gemmBlocks, 256, 0, stream>>>(
            hIn, bufAgg, eps, W1, b1, bufY, stats1, NNODES);
        bn_kernel<true><<<bnBlocks, 256, 0, stream>>>(bufY, stats1, g1, be1,
                                                      NNODES * FEAT, invN);

        // Y2 = act @ W2 + b2 (h no longer needed -> overwrite bufH).
        gemm_wmma_kernel<FEAT, false, true><<<gemmBlocks, 256, 0, stream>>>(
            bufY, nullptr, nullptr, W2, b2, bufH, stats2, NNODES);
        if (L != 2)
            bn_kernel<true><<<bnBlocks, 256, 0, stream>>>(bufH, stats2, g2, be2,
                                                          NNODES * FEAT, invN);
        else
            bn_kernel<false><<<bnBlocks, 256, 0, stream>>>(bufH, stats2, g2, be2,
                                                           NNODES * FEAT, invN);
        hIn = bufH;
    }

    // logits = h @ Wh + bh  -> first N*64 floats of d_out.
    const float* Wh = (const float*)d_in[30];
    const float* bh = (const float*)d_in[31];
    float* logits = (float*)d_out;
    gemm_wmma_kernel<OUTF, false, false><<<gemmBlocks, 256, 0, stream>>>(
        bufH, nullptr, nullptr, Wh, bh, logits, nullptr, NNODES);

    // Second output: final h, copied behind logits.
    hipMemcpyAsync((float*)d_out + (size_t)NNODES * OUTF, bufH,
                   (size_t)NNODES * FEAT * sizeof(float),
                   hipMemcpyDeviceToDevice, stream);
}